// DClusConv_49667001811500
// MI455X (gfx1250) — compile-verified
//
#include <hip/hip_runtime.h>

// ---- problem constants (from reference setup) ----
#define C_INP 128
#define C_MID 256          // 2*C_IN
#define GROUPS 4
#define CPG 64             // C_MID / GROUPS
#define KNN 9
#define BATCH 8
#define NPTS 3136          // 56*56
#define NT 196             // NPTS / 16
#define KCONV 576          // CPG * KNN

#define ROWS_WG 64         // rows per k_knn workgroup (4 waves x 16)
#define A_FLOATS (4 * 128 * 16 * 2)   // 16384 floats = 64 KB
#define B_FLOATS (128 * 16 * 2)       // 4096 floats = 16 KB per buffer
#define DT_STRIDE 20                  // padded row stride (16B-aligned float4 reads)

typedef float v2f __attribute__((ext_vector_type(2)));
typedef float v8f __attribute__((ext_vector_type(8)));

__device__ __forceinline__ v8f wmma_f32(v2f a, v2f b, v8f c) {
    // D = A(16x4,f32) * B(4x16,f32) + C(16x16,f32)
    return __builtin_amdgcn_wmma_f32_16x16x4_f32(false, a, false, b, (short)0, c,
                                                 false, false);
}

// low 32 bits of a flat shared-aperture address == wave-relative LDS byte offset
__device__ __forceinline__ unsigned lds_off(const void* p) {
    return (unsigned)(uintptr_t)p;
}
// ASYNCcnt-tracked DMA of one dword per lane: copies the dword at global address gaddr to LDS offset ldsoff
__device__ __forceinline__ void async_ld_b32(unsigned ldsoff, const float* gaddr) {
    asm volatile("global_load_async_to_lds_b32 %0, %1, off"
                 :: "v"(ldsoff), "v"(gaddr) : "memory");
}
__device__ __forceinline__ void wait_async_le32() {
    asm volatile("s_wait_asynccnt 0x20" ::: "memory");
}
__device__ __forceinline__ void wait_async_0() {
    asm volatile("s_wait_asynccnt 0x0" ::: "memory");
}

// ---------------------------------------------------------------------------
// Kernel 1: fc1 (M=256, K=128, N=3136 per batch) + fused StarReLU.
// One wave per 16x16 output tile.
// ---------------------------------------------------------------------------
__global__ void __launch_bounds__(32)
k_fc1(const float* __restrict__ x, const float* __restrict__ w1,
      const float* __restrict__ ss, const float* __restrict__ sb,
      float* __restrict__ h)
{
    const int nt   = blockIdx.x;
    const int ot   = blockIdx.y;
    const int b    = blockIdx.z;
    const int lane = threadIdx.x;
    const int half = lane >> 4;
    const int l16  = lane & 15;
    const int n0 = nt * 16, o0 = ot * 16;

    const float* xb   = x + (size_t)b * C_INP * NPTS;
    const float* wrow = w1 + (size_t)(o0 + l16) * C_INP + 2 * half;
    const float* bcol = xb + (size_t)(2 * half) * NPTS + (n0 + l16);

    v8f acc = {};
    #pragma unroll 4
    for (int k0 = 0; k0 < C_INP; k0 += 4) {
        v2f a, bb;
        a.x  = wrow[k0];
        a.y  = wrow[k0 + 1];
        bb.x = bcol[(size_t)k0 * NPTS];
        bb.y = bcol[(size_t)(k0 + 1) * NPTS];
        acc = wmma_f32(a, bb, acc);
    }
    const float sS = ss[0], sB = sb[0];
    float* hb = h + (size_t)b * C_MID * NPTS;
    #pragma unroll
    for (int r = 0; r < 8; ++r) {
        float v = acc[r];
        v = v > 0.f ? v : 0.f;
        v = sS * v * v + sB;
        hb[(size_t)(o0 + r + 8 * half) * NPTS + (n0 + l16)] = v;
    }
}

// ---------------------------------------------------------------------------
// Kernel 2: per-node squared norms (float2-vectorized)
// ---------------------------------------------------------------------------
__global__ void __launch_bounds__(256)
k_sq(const float* __restrict__ h, float* __restrict__ sq)
{
    int t = blockIdx.x * 256 + threadIdx.x;           // pair index
    if (t >= BATCH * (NPTS / 2)) return;
    int b = t / (NPTS / 2), n2 = t % (NPTS / 2);
    const float* hb = h + (size_t)b * C_MID * NPTS + 2 * n2;
    float ax = 0.f, ay = 0.f;
    #pragma unroll 8
    for (int c = 0; c < C_MID; ++c) {
        float2 v = *(const float2*)&hb[(size_t)c * NPTS];
        ax += v.x * v.x;
        ay += v.y * v.y;
    }
    *(float2*)&sq[(size_t)b * NPTS + 2 * n2] = make_float2(ax, ay);
}

// ---------------------------------------------------------------------------
// Kernel 3: Gram (WMMA f32, K=256) + neg-dist + fused stable top-9.
// 128 threads own 64 rows (one 16-row block per wave). A panels (64 KB) and
// double-buffered B tiles (2 x 16 KB) are staged via global_load_async_to_lds;
// stage(ct+1) is issued before compute(ct) and in-order async completion makes
// s_wait_asynccnt<=32 the "current buffer resident" condition.
// ---------------------------------------------------------------------------
__global__ void __launch_bounds__(128)
k_knn(const float* __restrict__ h, const float* __restrict__ sq,
      int* __restrict__ idxout)
{
    extern __shared__ char smem[];
    float* Apan = (float*)smem;                 // [wave][kpair 128][m 16][2]
    float* Bpan = Apan + A_FLOATS;              // [2][kpair 128][j 16][2]
    float* Dt   = Bpan + 2 * B_FLOATS;          // [wave][16][DT_STRIDE]

    const int wgx  = blockIdx.x;                // 0..48
    const int b    = blockIdx.y;
    const int tid  = threadIdx.x;
    const int wave = tid >> 5, lane = tid & 31;
    const int half = lane >> 4, l16 = lane & 15;
    const int r0   = wgx * ROWS_WG;

    const float* hb  = h  + (size_t)b * C_MID * NPTS;
    const float* sqb = sq + (size_t)b * NPTS;

    // async-stage A panels: A[w][k][m] = h[k][r0 + w*16 + m], pair-interleaved
    for (int e = tid; e < A_FLOATS; e += 128) {
        int k = e >> 6, row = e & 63;
        int w = row >> 4, m = row & 15;
        unsigned off = lds_off(&Apan[(((w << 7) + (k >> 1)) * 16 + m) * 2 + (k & 1)]);
        async_ld_b32(off, hb + (size_t)k * NPTS + (r0 + row));
    }
    // async-stage B tile ct into buffer d
    auto stageB = [&](int ct, int d) {
        const int c0 = ct * 16;
        for (int e = tid; e < B_FLOATS; e += 128) {
            int k = e >> 4, j = e & 15;
            unsigned off =
                lds_off(&Bpan[d * B_FLOATS + ((k >> 1) * 16 + j) * 2 + (k & 1)]);
            async_ld_b32(off, hb + (size_t)k * NPTS + (c0 + j));
        }
    };
    stageB(0, 0);

    float msq[8];
    #pragma unroll
    for (int r = 0; r < 8; ++r) msq[r] = sqb[r0 + wave * 16 + r + 8 * half];

    float vals[KNN];
    int   inds[KNN];
    #pragma unroll
    for (int i = 0; i < KNN; ++i) { vals[i] = -3.4e38f; inds[i] = 0; }

    const float* Aw = Apan + (wave << 12);      // wave*128*16*2
    float*       Dw = Dt + wave * (16 * DT_STRIDE);

    for (int ct = 0; ct < NT; ++ct) {
        const int d = ct & 1;
        if (ct + 1 < NT) { stageB(ct + 1, d ^ 1); wait_async_le32(); }
        else             { wait_async_0(); }
        __syncthreads();                        // buffer d resident for all waves

        const float* Bw = Bpan + d * B_FLOATS;
        v8f acc = {};
        #pragma unroll 8
        for (int k0 = 0; k0 < C_MID; k0 += 4) {
            int kp = (k0 >> 1) + half;
            float2 av = *(const float2*)&Aw[(kp * 16 + l16) * 2];
            float2 bv = *(const float2*)&Bw[(kp * 16 + l16) * 2];
            v2f a, bb;
            a.x  = av.x;  a.y  = av.y;
            bb.x = bv.x;  bb.y = bv.y;
            acc = wmma_f32(a, bb, acc);
        }

        const int   c0  = ct * 16;
        const float csq = sqb[c0 + l16];
        #pragma unroll
        for (int r = 0; r < 8; ++r)
            Dw[(r + 8 * half) * DT_STRIDE + l16] = 2.f * acc[r] - msq[r] - csq;

        // lanes 0..15 of each wave: one row each; ascending col order +
        // insert-after-equal == jax.lax.top_k stable tie rule
        if (lane < 16) {
            #pragma unroll
            for (int jj = 0; jj < 4; ++jj) {
                float4 q = *(const float4*)&Dw[lane * DT_STRIDE + jj * 4];
                float cand[4] = {q.x, q.y, q.z, q.w};
                #pragma unroll
                for (int u = 0; u < 4; ++u) {
                    float v   = cand[u];
                    int   col = c0 + jj * 4 + u;
                    if (v > vals[KNN - 1]) {
                        int rank = 0;
                        #pragma unroll
                        for (int i = 0; i < KNN; ++i) rank += (vals[i] >= v) ? 1 : 0;
                        #pragma unroll
                        for (int s = KNN - 1; s >= 0; --s) {
                            float pv = (s > 0) ? vals[s - 1] : v;
                            int   pi = (s > 0) ? inds[s - 1] : col;
                            bool place = (s == rank);
                            bool shift = (s > rank);
                            vals[s] = place ? v   : (shift ? pv : vals[s]);
                            inds[s] = place ? col : (shift ? pi : inds[s]);
                        }
                    }
                }
            }
        }
        __syncthreads();                        // all reads of buffer d done
    }

    if (lane < 16) {
        int* op = idxout + ((size_t)b * NPTS + (r0 + wave * 16 + lane)) * KNN;
        #pragma unroll
        for (int i = 0; i < KNN; ++i) op[i] = inds[i];
    }
}

// ---------------------------------------------------------------------------
// Kernel 4: async-gather neighbors into LDS + grouped conv as WMMA GEMM
// (per group: W[64 x 576] * G[576 x 16]); wave w computes out-subtile w.
// ---------------------------------------------------------------------------
__global__ void __launch_bounds__(128)
k_gconv(const float* __restrict__ h, const int* __restrict__ idx,
        const float* __restrict__ wc, float* __restrict__ y)
{
    __shared__ float G[KCONV / 2][16][2];   // pair-interleaved, 36 KB
    __shared__ int   nbr[16][KNN];

    const int nt  = blockIdx.x;
    const int b   = blockIdx.y;
    const int tid = threadIdx.x;
    const int n0  = nt * 16;
    const float* hb = h + (size_t)b * C_MID * NPTS;

    if (tid < 16 * KNN) {
        int j = tid / KNN, k = tid % KNN;
        nbr[j][k] = idx[((size_t)b * NPTS + n0 + j) * KNN + k];
    }
    __syncthreads();

    const int wave = tid >> 5;
    const int lane = tid & 31;
    const int half = lane >> 4;
    const int l16  = lane & 15;
    float* yb = y + (size_t)b * C_MID * NPTS;

    for (int g = 0; g < GROUPS; ++g) {
        // async random-gather: G[kidx][j] = h[g*64 + kidx/9][ nbr[j][kidx%9] ]
        for (int e = tid; e < KCONV * 16; e += 128) {
            int j    = e & 15;
            int kidx = e >> 4;
            int i    = kidx / KNN, k = kidx % KNN;
            unsigned off = lds_off(&G[kidx >> 1][j][kidx & 1]);
            async_ld_b32(off, hb + (size_t)(g * CPG + i) * NPTS + nbr[j][k]);
        }
        wait_async_0();
        __syncthreads();

        const int o0 = g * CPG + wave * 16;
        const float* wrow = wc + (size_t)(o0 + l16) * KCONV + 2 * half;
        v8f acc = {};
        #pragma unroll 4
        for (int k0 = 0; k0 < KCONV; k0 += 4) {
            v2f a, bb;
            a.x = wrow[k0];
            a.y = wrow[k0 + 1];
            float2 bv = *(const float2*)&G[(k0 >> 1) + half][l16][0];
            bb.x = bv.x;  bb.y = bv.y;
            acc = wmma_f32(a, bb, acc);
        }
        #pragma unroll
        for (int r = 0; r < 8; ++r)
            yb[(size_t)(o0 + r + 8 * half) * NPTS + (n0 + l16)] = acc[r];
        __syncthreads();
    }
}

// ---------------------------------------------------------------------------
// Kernel 5: fc2 over the reference's "faithful" scrambled reshape:
// permuted view == y_flat[n*256 + c2]  (pure index math, no transpose pass)
// ---------------------------------------------------------------------------
__global__ void __launch_bounds__(32)
k_fc2(const float* __restrict__ y, const float* __restrict__ w2,
      float* __restrict__ out)
{
    const int nt   = blockIdx.x;
    const int ot   = blockIdx.y;
    const int b    = blockIdx.z;
    const int lane = threadIdx.x;
    const int half = lane >> 4;
    const int l16  = lane & 15;
    const int n0 = nt * 16, o0 = ot * 16;

    const float* yb   = y + (size_t)b * C_MID * NPTS;
    const float* wrow = w2 + (size_t)(o0 + l16) * C_MID + 2 * half;
    const float* bcol = yb + (size_t)(n0 + l16) * C_MID + 2 * half;

    v8f acc = {};
    #pragma unroll 4
    for (int k0 = 0; k0 < C_MID; k0 += 4) {
        v2f a, bb;
        a.x  = wrow[k0];   a.y  = wrow[k0 + 1];
        bb.x = bcol[k0];   bb.y = bcol[k0 + 1];
        acc = wmma_f32(a, bb, acc);
    }
    float* ob = out + (size_t)b * C_INP * NPTS;
    #pragma unroll
    for (int r = 0; r < 8; ++r)
        ob[(size_t)(o0 + r + 8 * half) * NPTS + (n0 + l16)] = acc[r];
}

// ---------------------------------------------------------------------------
extern "C" void kernel_launch(void* const* d_in, const int* in_sizes, int n_in,
                              void* d_out, int out_size, void* d_ws, size_t ws_size,
                              hipStream_t stream)
{
    (void)in_sizes; (void)n_in; (void)out_size; (void)ws_size;
    const float* x  = (const float*)d_in[0];
    const float* w1 = (const float*)d_in[1];
    const float* ss = (const float*)d_in[2];
    const float* sb = (const float*)d_in[3];
    const float* wc = (const float*)d_in[4];
    const float* w2 = (const float*)d_in[5];
    float* out = (float*)d_out;

    char* ws = (char*)d_ws;
    size_t off = 0;
    float* h  = (float*)(ws + off); off += (size_t)BATCH * C_MID * NPTS * 4;  // 25.7 MB
    float* sq = (float*)(ws + off); off += (size_t)BATCH * NPTS * 4;          // 100 KB
    int*   ix = (int*)  (ws + off); off += (size_t)BATCH * NPTS * KNN * 4;    // 903 KB
    float* y  = (float*)(ws + off);                                           // 25.7 MB

    const size_t knn_lds = (size_t)(A_FLOATS + 2 * B_FLOATS + 4 * 16 * DT_STRIDE)
                           * sizeof(float);  // 103,424 B

    k_fc1 <<<dim3(NT, C_MID / 16, BATCH), 32, 0, stream>>>(x, w1, ss, sb, h);
    k_sq  <<<dim3((BATCH * (NPTS / 2) + 255) / 256), 256, 0, stream>>>(h, sq);
    k_knn <<<dim3(NPTS / ROWS_WG, BATCH), 128, knn_lds, stream>>>(h, sq, ix);
    k_gconv<<<dim3(NT, BATCH), 128, 0, stream>>>(h, ix, wc, y);
    k_fc2 <<<dim3(NT, C_INP / 16, BATCH), 32, 0, stream>>>(y, w2, out);
}